// NavierStokesLossMAC_79104707658144
// MI455X (gfx1250) — compile-verified
//
#include <hip/hip_runtime.h>
#include <hip/hip_bf16.h>
#include <cstdint>

// Problem constants (fixed by setup_inputs / reference)
#define HH   1024
#define WW   1024
#define NB   8
#define RHO_C  1.0f
#define MU_C   0.1f
#define INV_DT 0.25f   // 1/DT, DT=4.0

// Tiling
#define TW   64
#define TH   32
#define HTW  (TW + 2)          // 66
#define HTH  (TH + 2)          // 34
#define HN   (HTW * HTH)       // 2244
#define NTX  16                // ceil(1022/64)
#define NTY  32                // ceil(1022/32)
#define NTILES (NTX * NTY)     // 512 blocks per batch

#if defined(__AMDGCN__)
typedef __attribute__((address_space(1))) int gas_int;
typedef __attribute__((address_space(3))) int las_int;
typedef unsigned int u32x4 __attribute__((ext_vector_type(4)));
typedef int i32x8 __attribute__((ext_vector_type(8)));
typedef int i32x4 __attribute__((ext_vector_type(4)));
#endif

#if defined(__AMDGCN__) && __has_builtin(__builtin_amdgcn_tensor_load_to_lds)
#define HAVE_TDM 1
#else
#define HAVE_TDM 0
#endif

// ---------------------------------------------------------------------------
// TDM: one descriptor loads a full HTH x HTW f32 tile (row stride WW) into LDS.
// Tile is guaranteed fully inside the tensor by the caller (shifted origin).
// D# layout per CDNA5 ISA. This toolchain exposes
// the 6-arg builtin: (g0 v4u, g1 v8i, g2 v4i, g3 v4i, v8i, cpol).
// ---------------------------------------------------------------------------
#if HAVE_TDM
__device__ __forceinline__ void tdm_load_tile_f32(const float* gbase, float* ldsbase) {
  unsigned lds_off =
      (unsigned)(uintptr_t)(__attribute__((address_space(3))) float*)ldsbase;
  unsigned long long ga = (unsigned long long)(uintptr_t)gbase;

  u32x4 g0;
  g0[0] = 1u;                                   // count=1 (valid), user mode
  g0[1] = lds_off;                              // lds_addr (bytes)
  g0[2] = (unsigned)(ga & 0xffffffffu);         // global_addr[31:0]
  g0[3] = (unsigned)((ga >> 32) & 0x01ffffffu)  // global_addr[56:32]
          | (2u << 30);                         // type=2 ("image")

  i32x8 g1;
  g1[0] = (int)(2u << 16);                      // data_size=2 (4B); mask/pad off
  g1[1] = (int)(((unsigned)WW & 0xffffu) << 16);            // tensor_dim0[15:0] @ [63:48]
  g1[2] = (int)((((unsigned)WW >> 16) & 0xffffu)            // tensor_dim0[31:16]
                | (((unsigned)HH & 0xffffu) << 16));        // tensor_dim1[15:0]
  g1[3] = (int)((((unsigned)HH >> 16) & 0xffffu)            // tensor_dim1[31:16]
                | ((unsigned)HTW << 16));                   // tile_dim0 = 66
  g1[4] = (int)((unsigned)HTH);                 // tile_dim1 = 34; tile_dim2 = 0
  g1[5] = (int)WW;                              // tensor_dim0_stride[31:0] (elements)
  g1[6] = 0;                                    // stride0 hi | dim1_stride lo
  g1[7] = 0;                                    // dim1_stride hi

  i32x4 g2 = {0, 0, 0, 0};                      // dims 2/3 unused
  i32x4 g3 = {0, 0, 0, 0};                      // dims 3/4 unused
  i32x8 g4 = {0, 0, 0, 0, 0, 0, 0, 0};          // trailing group (unused, zero)

  __builtin_amdgcn_tensor_load_to_lds(g0, g1, g2, g3, g4, 0);
}

__device__ __forceinline__ void wait_tensor0() {
  __builtin_amdgcn_s_wait_tensorcnt(0);
}
#endif

// ---------------------------------------------------------------------------
// Fallback: per-lane async global->LDS copy (ASYNCcnt-tracked), probe-verified.
// ---------------------------------------------------------------------------
__device__ __forceinline__ void async_copy_f32(const float* gsrc, float* ldst) {
#if defined(__AMDGCN__) && __has_builtin(__builtin_amdgcn_global_load_async_to_lds_b32)
  __builtin_amdgcn_global_load_async_to_lds_b32(
      (gas_int*)gsrc, (las_int*)ldst, /*offset=*/0, /*cpol=*/0);
#elif defined(__AMDGCN__)
  unsigned lds_off =
      (unsigned)(uintptr_t)(__attribute__((address_space(3))) void*)ldst;
  asm volatile("global_load_async_to_lds_b32 %0, %1, off"
               :: "v"(lds_off), "v"(gsrc)
               : "memory");
#else
  *ldst = *gsrc;  // host-pass parse only
#endif
}

__device__ __forceinline__ void wait_async0() {
#if defined(__AMDGCN__) && __has_builtin(__builtin_amdgcn_s_wait_asynccnt)
  __builtin_amdgcn_s_wait_asynccnt(0);
#elif defined(__AMDGCN__)
  asm volatile("s_wait_asynccnt 0" ::: "memory");
#endif
}

#define LI(dy_, dx_) ((ly + (dy_)) * HTW + (lx + (dx_)))

__global__ __launch_bounds__(256) void ns_loss_main(
    const float* __restrict__ vold, const float* __restrict__ vnew,
    const float* __restrict__ pnew, const int* __restrict__ mask,
    float* __restrict__ partial) {
  __shared__ float s_uo[HN], s_un[HN], s_wo[HN], s_wn[HN], s_p[HN];
  __shared__ float s_red[8];

  const int tid = threadIdx.x;
  const int tX = blockIdx.x, tY = blockIdx.y, b = blockIdx.z;

  const size_t bV = (size_t)b * (2u * HH * WW);
  const float* uo_g = vold + bV;
  const float* wo_g = vold + bV + (size_t)HH * WW;
  const float* un_g = vnew + bV;
  const float* wn_g = vnew + bV + (size_t)HH * WW;
  const float* p_g  = pnew + (size_t)b * HH * WW;
  const int*   m_g  = mask + (size_t)b * HH * WW;

  // Halo window origin, shifted so the 66x34 window is fully in-bounds.
  const int gx0 = tX * TW;
  const int gy0 = tY * TH;
  const int ox = (gx0 > WW - HTW) ? (WW - HTW) : gx0;
  const int oy = (gy0 > HH - HTH) ? (HH - HTH) : gy0;
  const int dx = gx0 - ox;   // 0 except rightmost tiles (==2)
  const int dy = gy0 - oy;   // 0 except bottom tiles (==2)
  const size_t org = (size_t)oy * WW + ox;

  // ---- Stage halo tiles into LDS -----------------------------------------
#if HAVE_TDM
  if (tid < 32) {            // wave 0 issues 5 TDM descriptors
    tdm_load_tile_f32(uo_g + org, s_uo);
    tdm_load_tile_f32(un_g + org, s_un);
    tdm_load_tile_f32(wo_g + org, s_wo);
    tdm_load_tile_f32(wn_g + org, s_wn);
    tdm_load_tile_f32(p_g  + org, s_p);
    wait_tensor0();
  }
  __syncthreads();
#else
  for (int idx = tid; idx < HN; idx += 256) {
    int hy = idx / HTW;
    int hx = idx - hy * HTW;
    size_t off = org + (size_t)hy * WW + hx;
    async_copy_f32(uo_g + off, &s_uo[idx]);
    async_copy_f32(un_g + off, &s_un[idx]);
    async_copy_f32(wo_g + off, &s_wo[idx]);
    async_copy_f32(wn_g + off, &s_wn[idx]);
    async_copy_f32(p_g  + off, &s_p[idx]);
  }
  wait_async0();
  __syncthreads();
#endif

  // ---- Stencil + residual ------------------------------------------------
  const int lx = dx + (tid & 63) + 1;   // column within halo window
  const int r0 = (tid >> 6) * 8;        // 8 consecutive rows per thread
  const int x  = gx0 + (tid & 63) + 1;  // global x
  const bool xok = (x <= WW - 2);
  float acc = 0.0f;

#pragma unroll
  for (int i = 0; i < 8; ++i) {
    const int y  = gy0 + r0 + i + 1;    // global y
    const int ly = dy + r0 + i + 1;     // row within halo window
    if (xok && y <= HH - 2) {
      float uoc = s_uo[LI(0, 0)],  unc = s_un[LI(0, 0)];
      float uol = s_uo[LI(0, -1)], unl = s_un[LI(0, -1)];
      float uor = s_uo[LI(0, 1)],  unr = s_un[LI(0, 1)];
      float uou = s_uo[LI(-1, 0)], unu = s_un[LI(-1, 0)];
      float uod = s_uo[LI(1, 0)],  und = s_un[LI(1, 0)];
      float woc = s_wo[LI(0, 0)],  wnc = s_wn[LI(0, 0)];
      float wol = s_wo[LI(0, -1)], wnl = s_wn[LI(0, -1)];
      float wor = s_wo[LI(0, 1)],  wnr = s_wn[LI(0, 1)];
      float wou = s_wo[LI(-1, 0)], wnu = s_wn[LI(-1, 0)];
      float wod = s_wo[LI(1, 0)],  wnd = s_wn[LI(1, 0)];
      float pc  = s_p[LI(0, 0)];
      float pxm = s_p[LI(0, -1)];
      float pym = s_p[LI(-1, 0)];

      float uc = 0.5f * (uoc + unc), ul = 0.5f * (uol + unl);
      float ur = 0.5f * (uor + unr), uu = 0.5f * (uou + unu);
      float ud = 0.5f * (uod + und);
      float wc = 0.5f * (woc + wnc), wl = 0.5f * (wol + wnl);
      float wr = 0.5f * (wor + wnr), wu = 0.5f * (wou + wnu);
      float wd = 0.5f * (wod + wnd);

      float m = (float)m_g[(size_t)y * WW + x];
      float dwdt = (wnc - woc) * INV_DT;
      float dudt = (unc - uoc) * INV_DT;

      // res_x: momentum in w with advection by u along y
      float conv_x = wc * 0.5f * (wr - wl) +
                     0.5f * ((0.5f * (uc + ul)) * (wc - wu) +
                             (0.5f * (uc + ur)) * (wd - wc));
      float lapw = wl + wr + wu + wd - 4.0f * wc;
      float rx = m * (RHO_C * (dwdt + conv_x) + (pc - pxm) - MU_C * lapw);

      // res_y: momentum in u with advection by w along x
      float conv_y = uc * 0.5f * (ud - uu) +
                     0.5f * ((0.5f * (wc + wu)) * (uc - ul) +
                             (0.5f * (wc + wd)) * (ur - uc));
      float lapu = ul + ur + uu + ud - 4.0f * uc;
      float ry = m * (RHO_C * (dudt + conv_y) + (pc - pym) - MU_C * lapu);

      acc += rx * rx + ry * ry;
    }
  }

  // ---- deterministic block reduction (wave32 shuffles) -------------------
#pragma unroll
  for (int off = 16; off > 0; off >>= 1) acc += __shfl_down(acc, off, 32);
  if ((tid & 31) == 0) s_red[tid >> 5] = acc;
  __syncthreads();
  if (tid == 0) {
    float s = 0.0f;
#pragma unroll
    for (int i = 0; i < 8; ++i) s += s_red[i];
    partial[(size_t)b * NTILES + tY * NTX + tX] = s;
  }
}

__global__ __launch_bounds__(256) void ns_loss_reduce(
    const float* __restrict__ partial, float* __restrict__ out) {
  __shared__ float s_red[8];
  const int b = blockIdx.x, tid = threadIdx.x;
  float acc = 0.0f;
  for (int i = tid; i < NTILES; i += 256) acc += partial[(size_t)b * NTILES + i];
#pragma unroll
  for (int off = 16; off > 0; off >>= 1) acc += __shfl_down(acc, off, 32);
  if ((tid & 31) == 0) s_red[tid >> 5] = acc;
  __syncthreads();
  if (tid == 0) {
    float s = 0.0f;
#pragma unroll
    for (int i = 0; i < 8; ++i) s += s_red[i];
    out[b] = s * (1.0f / ((float)(HH - 2) * (float)(WW - 2)));
  }
}

extern "C" void kernel_launch(void* const* d_in, const int* in_sizes, int n_in,
                              void* d_out, int out_size, void* d_ws, size_t ws_size,
                              hipStream_t stream) {
  const float* vold = (const float*)d_in[0];
  const float* vnew = (const float*)d_in[1];
  const float* pnew = (const float*)d_in[2];
  const int*   msk  = (const int*)d_in[3];
  float* partial = (float*)d_ws;  // NB*NTILES floats = 16 KB, fully overwritten

  dim3 grid(NTX, NTY, NB);
  ns_loss_main<<<grid, 256, 0, stream>>>(vold, vnew, pnew, msk, partial);
  ns_loss_reduce<<<NB, 256, 0, stream>>>(partial, (float*)d_out);
}